// KNN_84774064488936
// MI455X (gfx1250) — compile-verified
//
#include <hip/hip_runtime.h>
#include <hip/hip_bf16.h>

typedef float v2f __attribute__((ext_vector_type(2)));
typedef float v8f __attribute__((ext_vector_type(8)));
typedef int   v4i __attribute__((ext_vector_type(4)));

#define AS1 __attribute__((address_space(1)))
#define AS3 __attribute__((address_space(3)))

#ifndef __has_builtin
#define __has_builtin(x) 0
#endif

// ---- CDNA5 async global->LDS copy (ASYNCcnt path) -------------------------
// Probe-confirmed: __builtin_amdgcn_global_load_async_to_lds_b128 exists with
// param 1 = 'v4i __device__ *' (AS1, non-const). 4 args: (gsrc, ldst, off, cpol).
__device__ __forceinline__ void async_cp_b128(const float* g, float* l) {
#if __has_builtin(__builtin_amdgcn_global_load_async_to_lds_b128)
  __builtin_amdgcn_global_load_async_to_lds_b128(
      (AS1 v4i*)(size_t)g, (AS3 v4i*)l, 0, 0);
#else
  unsigned l32 = (unsigned)(unsigned long long)(AS3 char*)l;
  unsigned long long g64 = (unsigned long long)g;
  asm volatile("global_load_async_to_lds_b128 %0, %1, off"
               :: "v"(l32), "v"(g64) : "memory");
#endif
}

__device__ __forceinline__ void wait_async_le8() {
#if __has_builtin(__builtin_amdgcn_s_wait_asynccnt)
  __builtin_amdgcn_s_wait_asynccnt(8);
#else
  asm volatile("s_wait_asynccnt 0x8" ::: "memory");
#endif
  asm volatile("" ::: "memory");
}

__device__ __forceinline__ void wait_async_le0() {
#if __has_builtin(__builtin_amdgcn_s_wait_asynccnt)
  __builtin_amdgcn_s_wait_asynccnt(0);
#else
  asm volatile("s_wait_asynccnt 0x0" ::: "memory");
#endif
  asm volatile("" ::: "memory");
}

#define TOP3_INSERT(dd, ii)                                   \
  do {                                                        \
    float _d = (dd); int _i = (ii);                           \
    if (_d < d2) {                                            \
      if (_d < d1) {                                          \
        d2 = d1; i2 = i1;                                     \
        if (_d < d0) { d1 = d0; i1 = i0; d0 = _d; i0 = _i; }  \
        else         { d1 = _d; i1 = _i; }                    \
      } else { d2 = _d; i2 = _i; }                            \
    }                                                         \
  } while (0)

// Per-wave quarter-tile staging: 16 rows x 64 cols, padded row stride of 68
// dwords in LDS (bank = (4*m + col) % 64 -> conflict-free A reads).
#define QROWS   16
#define QCOLS   64
#define QSTRIDE 68                       // floats per row in LDS
#define QFLOATS (QROWS * QSTRIDE)        // 1088 floats = 4352 B per buffer
#define WAVES_PER_BLK 8
#define SMEM_FLOATS (WAVES_PER_BLK * 2 * QFLOATS)

// Issue one quarter (8 async b128; each moves two coalesced 256B row-segments)
__device__ __forceinline__ void issue_quarter(const float* __restrict__ T,
                                              int rowBase, int qq,
                                              float* __restrict__ buf, int lane) {
  const int r0  = lane >> 4;     // 0 or 1: which of the two rows this lane serves
  const int c16 = lane & 15;     // 16B slice within the 256B row segment
  const float* g = T + (size_t)(rowBase + r0) * 256 + qq * QCOLS + c16 * 4;
  float* l = buf + r0 * QSTRIDE + c16 * 4;
#pragma unroll
  for (int i = 0; i < 8; ++i) {
    async_cp_b128(g + (size_t)i * 512, l + i * (2 * QSTRIDE));
  }
}

// Kernel 1: stream train_data through LDS (async double-buffer), dot products
// via V_WMMA_F32_16X16X4_F32, per-lane top-3, block reduce to workspace.
__global__ __launch_bounds__(256) void knn_wmma_topk(
    const float* __restrict__ x,
    const float* __restrict__ T,
    int N, int numTiles,
    float* __restrict__ wsD, int* __restrict__ wsI)
{
  extern __shared__ float smem[];                 // double buffers, per wave
  __shared__ float xs[256];
  __shared__ float sD[WAVES_PER_BLK * 16 * 3];
  __shared__ int   sI[WAVES_PER_BLK * 16 * 3];

  const int tid = threadIdx.x;
  xs[tid] = x[tid];                               // D == 256
  __syncthreads();

  const int lane = tid & 31;
  const int w    = tid >> 5;
  const int gw   = blockIdx.x * WAVES_PER_BLK + w;
  const int gwN  = gridDim.x * WAVES_PER_BLK;

  const int h    = lane >> 4;                     // K sub-pair {0,1} vs {2,3}
  const int m    = lane & 15;                     // row-in-tile (A operand)
  const int koff = h << 1;

  float* buf[2];
  buf[0] = smem + (size_t)w * 2 * QFLOATS;
  buf[1] = buf[0] + QFLOATS;

  const float FINF = __builtin_inff();
  float d0 = FINF, d1 = FINF, d2 = FINF;
  int   i0 = -1,   i1 = -1,   i2 = -1;

  int parity = 0;
  if (gw < numTiles) issue_quarter(T, gw * 16, 0, buf[0], lane);

  for (int t = gw; t < numTiles; t += gwN) {
    v8f acc0 = {};                                // two accumulator chains
    v8f acc1 = {};
    float nrm = 0.0f;
    const int row = t * 16 + m;

#pragma unroll
    for (int qq = 0; qq < 4; ++qq) {
      // issue next quarter into the other buffer, then wait for current
      int tn, qn;
      if (qq < 3) { tn = t;       qn = qq + 1; }
      else        { tn = t + gwN; qn = 0;      }
      if (tn < numTiles) {
        issue_quarter(T, tn * 16, qn, buf[parity ^ 1], lane);
        wait_async_le8();
      } else {
        wait_async_le0();
      }

      const float* arow = buf[parity] + m * QSTRIDE + koff;  // conflict-free
      const float* xq   = xs + qq * QCOLS + koff;
#pragma unroll
      for (int c = 0; c < 16; c += 2) {
        v2f A0 = *(const v2f*)(arow + c * 4);
        v2f A1 = *(const v2f*)(arow + c * 4 + 4);
        v2f B0 = *(const v2f*)(xq + c * 4);
        v2f B1 = *(const v2f*)(xq + c * 4 + 4);
        acc0 = __builtin_amdgcn_wmma_f32_16x16x4_f32(
            false, A0, false, B0, (short)0, acc0, false, false);
        acc1 = __builtin_amdgcn_wmma_f32_16x16x4_f32(
            false, A1, false, B1, (short)0, acc1, false, false);
        nrm += A0.x * A0.x + A0.y * A0.y + A1.x * A1.x + A1.y * A1.y;
      }
      parity ^= 1;
    }

    v8f cs = acc0 + acc1;

    // full ||t||^2: lanes l and l^16 hold complementary K halves of row (l&15)
    nrm += __shfl_xor(nrm, 16, 32);

    // extract this lane's dot from the replicated C columns:
    // lanes 0-15 hold rows 0-7 in cs[0..7]; lanes 16-31 hold rows 8-15.
    const int j = lane & 7;
    float e = cs[0];
    e = (j == 1) ? cs[1] : e;
    e = (j == 2) ? cs[2] : e;
    e = (j == 3) ? cs[3] : e;
    e = (j == 4) ? cs[4] : e;
    e = (j == 5) ? cs[5] : e;
    e = (j == 6) ? cs[6] : e;
    e = (j == 7) ? cs[7] : e;
    const int src = (lane < 8) ? lane : lane + 8;   // lane l<16 owns row l
    float dot = __shfl(e, src, 32);

    float score = (lane < 16 && row < N) ? (nrm - 2.0f * dot) : FINF;
    TOP3_INSERT(score, row);
  }

  if (lane < 16) {
    const int b = (w * 16 + m) * 3;
    sD[b + 0] = d0; sI[b + 0] = i0;
    sD[b + 1] = d1; sI[b + 1] = i1;
    sD[b + 2] = d2; sI[b + 2] = i2;
  }
  __syncthreads();

  if (tid == 0) {
    float e0 = FINF, e1 = FINF, e2 = FINF;
    int   j0 = -1,   j1 = -1,   j2 = -1;
    for (int q = 0; q < WAVES_PER_BLK * 16 * 3; ++q) {
      float _d = sD[q]; int _i = sI[q];
      if (_d < e2) {
        if (_d < e1) {
          e2 = e1; j2 = j1;
          if (_d < e0) { e1 = e0; j1 = j0; e0 = _d; j0 = _i; }
          else         { e1 = _d; j1 = _i; }
        } else { e2 = _d; j2 = _i; }
      }
    }
    const int o = blockIdx.x * 3;
    wsD[o + 0] = e0; wsI[o + 0] = j0;
    wsD[o + 1] = e1; wsI[o + 1] = j1;
    wsD[o + 2] = e2; wsI[o + 2] = j2;
  }
}

// Kernel 2: merge per-block candidates; mode of the 3 nearest targets
// (most frequent; smallest value on ties).
__global__ __launch_bounds__(128) void knn_finalize(
    const float* __restrict__ wsD, const int* __restrict__ wsI,
    const int* __restrict__ tgt, int nCand, float* __restrict__ out)
{
  __shared__ float sD[128 * 3];
  __shared__ int   sI[128 * 3];
  const int tid = threadIdx.x;
  const float FINF = __builtin_inff();

  float d0 = FINF, d1 = FINF, d2 = FINF;
  int   i0 = -1,   i1 = -1,   i2 = -1;
  for (int q = tid; q < nCand; q += 128) TOP3_INSERT(wsD[q], wsI[q]);

  sD[tid * 3 + 0] = d0; sI[tid * 3 + 0] = i0;
  sD[tid * 3 + 1] = d1; sI[tid * 3 + 1] = i1;
  sD[tid * 3 + 2] = d2; sI[tid * 3 + 2] = i2;
  __syncthreads();

  if (tid == 0) {
    float e0 = FINF, e1 = FINF, e2 = FINF;
    int   j0 = -1,   j1 = -1,   j2 = -1;
    for (int q = 0; q < 128 * 3; ++q) {
      float _d = sD[q]; int _i = sI[q];
      if (_d < e2) {
        if (_d < e1) {
          e2 = e1; j2 = j1;
          if (_d < e0) { e1 = e0; j1 = j0; e0 = _d; j0 = _i; }
          else         { e1 = _d; j1 = _i; }
        } else { e2 = _d; j2 = _i; }
      }
    }
    int t0 = (j0 >= 0) ? tgt[j0] : 0;
    int t1 = (j1 >= 0) ? tgt[j1] : t0;
    int t2 = (j2 >= 0) ? tgt[j2] : t0;

    int n0 = 1 + (t0 == t1) + (t0 == t2);
    int n1 = 1 + (t1 == t0) + (t1 == t2);
    int n2 = 1 + (t2 == t0) + (t2 == t1);
    int mx = n0 > n1 ? n0 : n1; mx = mx > n2 ? mx : n2;
    int best = 0x7fffffff;
    if (n0 == mx && t0 < best) best = t0;
    if (n1 == mx && t1 < best) best = t1;
    if (n2 == mx && t2 < best) best = t2;
    out[0] = (float)best;
  }
}

extern "C" void kernel_launch(void* const* d_in, const int* in_sizes, int n_in,
                              void* d_out, int out_size, void* d_ws, size_t ws_size,
                              hipStream_t stream) {
  const float* x   = (const float*)d_in[0];   // [256]
  const float* T   = (const float*)d_in[1];   // [N, 256]
  const int*   tgt = (const int*)d_in[2];     // [N]
  const int N = in_sizes[2];
  const int numTiles = (N + 15) / 16;

  int NB = 512;
  const size_t perBlk = 3 * (sizeof(float) + sizeof(int));
  if (ws_size < (size_t)NB * perBlk) {
    NB = (int)(ws_size / perBlk);
    if (NB < 1) NB = 1;
  }

  float* wsD = (float*)d_ws;
  int*   wsI = (int*)(wsD + (size_t)NB * 3);

  const size_t smemBytes = (size_t)SMEM_FLOATS * sizeof(float);  // 69,632 B
  knn_wmma_topk<<<NB, 256, smemBytes, stream>>>(x, T, N, numTiles, wsD, wsI);
  knn_finalize<<<1, 128, 0, stream>>>(wsD, wsI, tgt, NB * 3, (float*)d_out);
}